// MaskRCNN_24670292148517
// MI455X (gfx1250) — compile-verified
//
#include <hip/hip_runtime.h>
#include <math.h>

// ---------------------------------------------------------------------------
// pyramid_roi_align as a per-box structured GEMM on CDNA5 WMMA (fp32, K=4).
//
//   Out[c, n] = sum_k Patch[c, k] * B[k, n]
//     Patch : 256 channels x 196 touched source pixels (14 rows x 14 cols)
//     B     : 196 x 49(->64) per-box bilinear weight matrix (4 nz / column)
//
// K = 196 = 49 * 4  ->  49 steps of V_WMMA_F32_16X16X4_F32 per 16x16 tile.
//  * A fragment: each lane's K pair (k, k+1) is an (x0, x1) bilinear pair;
//    after re-anchoring the right-edge clamp (exact: displaced weight is 0)
//    every pair is CONTIGUOUS -> one global_load_b64 per lane per K-step.
//  * B fragment: stored in LDS pre-paired as sBp[kp][n] = {B[2kp][n],
//    B[2kp+1][n]} -> one aligned ds_load_b64 lands directly in the even
//    VGPR pair WMMA wants (no shuffle movs). Pair-row stride 160 dwords
//    == 32 (mod 64) -> the two 16-lane halves hit disjoint LDS banks.
//  * B has only 196 nonzeros: vector zero-fill + 196 scalar writes.
// ---------------------------------------------------------------------------

typedef __attribute__((ext_vector_type(2))) float v2f;
typedef __attribute__((ext_vector_type(4))) float v4f;
typedef __attribute__((ext_vector_type(8))) float v8f;

#define POOL     7
#define NSRC     14            // 2*POOL source rows (and cols) per box
#define KDIM     196           // 14*14 touched source pixels = 49 * 4
#define KSTEPS   49            // KDIM / 4 (one per v_wmma_f32_16x16x4_f32)
#define NPAIR    98            // KDIM / 2 contiguous (x0,x1) K pairs
#define NPAD     64            // 49 outputs padded to 4 WMMA N-tiles
#define PSTRIDE2 160           // dwords per pair-row (2*NPAD padded to 160)
#define CCH      256           // channels
#define NTHREADS 512           // 16 waves -> 16 M-tiles of 16 channels

__global__ __launch_bounds__(NTHREADS, 1)
void roi_align_wmma_kernel(const float* __restrict__ boxes,
                           const float* __restrict__ p2,
                           const float* __restrict__ p3,
                           const float* __restrict__ p4,
                           const float* __restrict__ p5,
                           float* __restrict__ out)
{
    __shared__ __align__(16) float sBp[NPAIR * PSTRIDE2]; // 62,720 B
    __shared__ int   sOffP[NPAIR];         // y*W + x0 per contiguous pair
    __shared__ int   sRow[NSRC];
    __shared__ int   sCol[NSRC];
    __shared__ float sWy[NSRC];            // includes vy validity
    __shared__ float sWx[NSRC];            // includes vx validity

    const int box = blockIdx.x;
    const int tid = threadIdx.x;

    const float by1 = boxes[box * 4 + 0];
    const float bx1 = boxes[box * 4 + 1];
    const float by2 = boxes[box * 4 + 2];
    const float bx2 = boxes[box * 4 + 3];

    // FPN level routing: 4 + log2(sqrt(h*w) / (224/1024)), round-half-even,
    // clip to [2,5]  (rintf == jnp.round under default RNE mode).
    const float bh = by2 - by1;
    const float bw = bx2 - bx1;
    int lvl = (int)rintf(4.0f + log2f(sqrtf(bh * bw) * (1024.0f / 224.0f)));
    lvl = lvl < 2 ? 2 : (lvl > 5 ? 5 : lvl);

    const float* fm;
    int H;
    switch (lvl) {
        case 2:  fm = p2; H = 256; break;
        case 3:  fm = p3; H = 128; break;
        case 4:  fm = p4; H = 64;  break;
        default: fm = p5; H = 32;  break;
    }
    const int W = H;

    // --- phase 1: sample coords (threads 0..13) + vector zero-fill of B
    if (tid < 2 * POOL) {
        const bool isY = tid < POOL;
        const int  i   = isY ? tid : tid - POOL;
        const float lo = isY ? by1 : bx1;
        const float hi = isY ? by2 : bx2;
        const float g  = (float)i * (1.0f / (float)(POOL - 1));
        const float s  = (lo + g * (hi - lo)) * (float)(H - 1);
        const float f0 = floorf(s);
        const float fr = s - f0;
        int i0 = (int)f0;
        i0 = i0 < 0 ? 0 : (i0 > H - 1 ? H - 1 : i0);
        const int i1 = (i0 + 1 > H - 1) ? H - 1 : i0 + 1;
        const float v = (s >= 0.0f && s <= (float)(H - 1)) ? 1.0f : 0.0f;
        float w0 = (1.0f - fr) * v;
        float w1 = fr * v;
        if (isY) {
            sRow[2 * i] = i0;  sRow[2 * i + 1] = i1;
            sWy[2 * i] = w0;   sWy[2 * i + 1] = w1;
        } else {
            int c0 = i0, c1 = i1;
            if (c1 == c0) {
                // Right-edge clamp: TF semantics force fr==0 or v==0 here,
                // so the combined weight w0+w1 belongs to element c0 and the
                // displaced slot is exactly zero. Re-anchor so the pair is
                // contiguous: (c0-1, c0) with weights (0, w0+w1).
                c0 = c1 - 1;
                w1 = w0 + w1;
                w0 = 0.0f;
            }
            sCol[2 * i] = c0;  sCol[2 * i + 1] = c1;
            sWx[2 * i] = w0;   sWx[2 * i + 1] = w1;
        }
    }
    for (int idx = tid * 4; idx < NPAIR * PSTRIDE2; idx += NTHREADS * 4)
        *(v4f*)&sBp[idx] = (v4f){0.0f, 0.0f, 0.0f, 0.0f};
    __syncthreads();

    // --- phase 2: 196 sparse weights + 98 pair-base offsets
    if (tid < KDIM) {
        const int k = tid;
        const int r = k / NSRC;
        const int s = k - r * NSRC;
        const int n = (r >> 1) * POOL + (s >> 1);   // the one nonzero column
        sBp[(k >> 1) * PSTRIDE2 + 2 * n + (k & 1)] = sWy[r] * sWx[s];
    }
    if (tid < NPAIR) {
        const int r = tid / (NSRC / 2);             // tid / 7
        const int s = 2 * (tid - r * (NSRC / 2));
        sOffP[tid] = sRow[r] * W + sCol[s];         // sCol[s+1]==sCol[s]+1
    }
    __syncthreads();

    // --- phase 3: per-wave 16(M) x 64(N) tile, 49 x (K=4) fp32 WMMAs.
    // A layout (16x4 f32): lane<16 -> M=lane,    K = k0,k0+1 (VGPR0,1)
    //                      lane>=16 -> M=lane-16, K = k0+2,k0+3
    // B layout (4x16 f32): VGPR0 -> K=k0 (lanes 0-15) / K=k0+2 (lanes 16-31)
    // One index kp = 2*kk + half drives BOTH the gather pair and the B pair.
    const int wave = tid >> 5;
    const int lane = tid & 31;
    const int half = lane >> 4;
    const int ml   = lane & 15;
    const int c0   = wave * 16;

    const float* __restrict__ fbase =
        fm + (size_t)(c0 + ml) * (size_t)(H * W);

    v8f acc0 = {}, acc1 = {}, acc2 = {}, acc3 = {};

    for (int kk = 0; kk < KSTEPS; ++kk) {
        const int kp = 2 * kk + half;

        const float* __restrict__ ap = fbase + sOffP[kp];
        v2f A;
        A.x = ap[0];                       // one address, two adjacent
        A.y = ap[1];                       // elements -> global_load_b64

        const float* bp = &sBp[kp * PSTRIDE2 + 2 * ml];
        v2f B0 = *(const v2f*)(bp);        // aligned ds_load_b64, lands
        v2f B1 = *(const v2f*)(bp + 32);   // directly in the WMMA operand
        v2f B2 = *(const v2f*)(bp + 64);   // register pair
        v2f B3 = *(const v2f*)(bp + 96);

        acc0 = __builtin_amdgcn_wmma_f32_16x16x4_f32(false, A, false, B0,
                                                     (short)0, acc0, false, false);
        acc1 = __builtin_amdgcn_wmma_f32_16x16x4_f32(false, A, false, B1,
                                                     (short)0, acc1, false, false);
        acc2 = __builtin_amdgcn_wmma_f32_16x16x4_f32(false, A, false, B2,
                                                     (short)0, acc2, false, false);
        acc3 = __builtin_amdgcn_wmma_f32_16x16x4_f32(false, A, false, B3,
                                                     (short)0, acc3, false, false);
    }

    // --- C/D layout: VGPR v -> M = v + 8*half, N = ml + 16*ntile
    float* __restrict__ obase =
        out + (size_t)box * (size_t)CCH * (POOL * POOL);
    #pragma unroll
    for (int v = 0; v < 8; ++v) {
        const int c = c0 + half * 8 + v;
        float vals[4] = { acc0[v], acc1[v], acc2[v], acc3[v] };
        #pragma unroll
        for (int nt = 0; nt < 4; ++nt) {
            const int n = nt * 16 + ml;
            if (n < POOL * POOL)
                obase[(size_t)c * (POOL * POOL) + n] = vals[nt];
        }
    }
}

extern "C" void kernel_launch(void* const* d_in, const int* in_sizes, int n_in,
                              void* d_out, int out_size, void* d_ws, size_t ws_size,
                              hipStream_t stream) {
    const float* boxes = (const float*)d_in[0];
    const float* p2    = (const float*)d_in[1];
    const float* p3    = (const float*)d_in[2];
    const float* p4    = (const float*)d_in[3];
    const float* p5    = (const float*)d_in[4];
    float* out = (float*)d_out;

    const int n_boxes = in_sizes[0] / 4;   // boxes: [N, 4]
    roi_align_wmma_kernel<<<n_boxes, NTHREADS, 0, stream>>>(
        boxes, p2, p3, p4, p5, out);
}